// GraphAttentionNetwork_53291954208987
// MI455X (gfx1250) — compile-verified
//
#include <hip/hip_runtime.h>
#include <hip/hip_bf16.h>

typedef __attribute__((ext_vector_type(2))) float v2f;
typedef __attribute__((ext_vector_type(8))) float v8f;

#define NEG_SLOPE 0.2f
#define C128 128
#define HID 32

static __device__ __forceinline__ float lrelu(float x) { return x >= 0.f ? x : NEG_SLOPE * x; }

// order-preserving float <-> uint map so unsigned atomicMax == float max
static __device__ __forceinline__ unsigned ford(float f) {
    unsigned u = __float_as_uint(f);
    return (u & 0x80000000u) ? ~u : (u | 0x80000000u);
}
static __device__ __forceinline__ float unford(unsigned u) {
    return (u & 0x80000000u) ? __uint_as_float(u ^ 0x80000000u) : __uint_as_float(~u);
}

// ---------------------------------------------------------------------------
// GEMM: H[N,128] = X[N,K] @ W[K,128]   (fp32 WMMA 16x16x4)
// block = 256 threads = 8 waves; wave w computes rows [16*bx,16*bx+16) x cols [16w,16w+16)
// ---------------------------------------------------------------------------
__global__ __launch_bounds__(256) void gat_gemm128_wmma(
    const float* __restrict__ X, const float* __restrict__ W,
    float* __restrict__ H, int N, int K) {
    const int wave = threadIdx.x >> 5;
    const int lane = threadIdx.x & 31;
    const int half = lane >> 4;    // 0: K pair {0,1}, 1: K pair {2,3}
    const int l16  = lane & 15;
    const int row0 = blockIdx.x << 4;
    const int col  = (wave << 4) + l16;

    int arow = row0 + l16;
    if (arow > N - 1) arow = N - 1;          // clamp loads, keep EXEC all-ones
    const float* xrow = X + (size_t)arow * K;

    v8f acc = {};
    for (int k0 = 0; k0 < K; k0 += 4) {
        const int ka = k0 + (half << 1);
        const float2 av = *(const float2*)(xrow + ka);
        v2f a; a.x = av.x; a.y = av.y;       // A[M=l16][ka], A[M=l16][ka+1]
        v2f b;
        b.x = W[(size_t)ka * C128 + col];    // B[ka][col]
        b.y = W[(size_t)(ka + 1) * C128 + col];
        acc = __builtin_amdgcn_wmma_f32_16x16x4_f32(false, a, false, b,
                                                    (short)0, acc, false, false);
    }
    // C/D layout: lane<16 -> M = j, lane>=16 -> M = j+8 ; N = l16
    const int rbase = row0 + (half << 3);
#pragma unroll
    for (int j = 0; j < 8; ++j) {
        const int r = rbase + j;
        if (r < N) H[(size_t)r * C128 + col] = acc[j];
    }
}

// ---------------------------------------------------------------------------
// attention logits, 4 heads x 32 dims: als[n,h] = <h[n,h,:], a_src[h,:]>
// ---------------------------------------------------------------------------
__global__ void attn_logits4(const float* __restrict__ H,
                             const float* __restrict__ a_src,
                             const float* __restrict__ a_dst,
                             float* __restrict__ als, float* __restrict__ ald, int N) {
    const int idx = blockIdx.x * blockDim.x + threadIdx.x; // n*4 + head
    if (idx >= N * 4) return;
    const int n = idx >> 2, hh = idx & 3;
    const float* hp  = H + (size_t)n * C128 + hh * HID;
    const float* asr = a_src + hh * HID;
    const float* adr = a_dst + hh * HID;
    float s0 = 0.f, s1 = 0.f;
#pragma unroll
    for (int d = 0; d < HID; ++d) { const float v = hp[d]; s0 += v * asr[d]; s1 += v * adr[d]; }
    als[idx] = s0; ald[idx] = s1;
}

static __device__ __forceinline__ void edge_sd(const int* __restrict__ src,
                                               const int* __restrict__ dst,
                                               int e, int E, int& s, int& d) {
    if (e < E) { s = src[e]; d = dst[e]; } else { s = e - E; d = s; }   // self-loops appended
}

// ---------------------------------------------------------------------------
// edge pass 1 (H=4): segment max of leakyrelu(als[src]+ald[dst]) into mmax (ford'd)
// ---------------------------------------------------------------------------
__global__ void edge_max4(const int* __restrict__ src, const int* __restrict__ dst,
                          const float* __restrict__ als, const float* __restrict__ ald,
                          unsigned* __restrict__ mmax, int E, int Et) {
    const int e = blockIdx.x * blockDim.x + threadIdx.x;
    if (e >= Et) return;
    int s, d; edge_sd(src, dst, e, E, s, d);
    const float4 a = *(const float4*)(als + 4 * s);
    const float4 b = *(const float4*)(ald + 4 * d);
    atomicMax(&mmax[4 * d + 0], ford(lrelu(a.x + b.x)));
    atomicMax(&mmax[4 * d + 1], ford(lrelu(a.y + b.y)));
    atomicMax(&mmax[4 * d + 2], ford(lrelu(a.z + b.z)));
    atomicMax(&mmax[4 * d + 3], ford(lrelu(a.w + b.w)));
}

// edge pass 2 (H=4): ssum[dst] += exp(e - m[dst])
__global__ void edge_sum4(const int* __restrict__ src, const int* __restrict__ dst,
                          const float* __restrict__ als, const float* __restrict__ ald,
                          const unsigned* __restrict__ mmax, float* __restrict__ ssum,
                          int E, int Et) {
    const int e = blockIdx.x * blockDim.x + threadIdx.x;
    if (e >= Et) return;
    int s, d; edge_sd(src, dst, e, E, s, d);
    const float4 a = *(const float4*)(als + 4 * s);
    const float4 b = *(const float4*)(ald + 4 * d);
    const uint4 mu = *(const uint4*)(mmax + 4 * d);
    atomicAdd(&ssum[4 * d + 0], expf(lrelu(a.x + b.x) - unford(mu.x)));
    atomicAdd(&ssum[4 * d + 1], expf(lrelu(a.y + b.y) - unford(mu.y)));
    atomicAdd(&ssum[4 * d + 2], expf(lrelu(a.z + b.z) - unford(mu.z)));
    atomicAdd(&ssum[4 * d + 3], expf(lrelu(a.w + b.w) - unford(mu.w)));
}

// edge pass 3 (H=4): agg[dst,:] += alpha[head(c)] * h[src,:]; one wave per edge,
// lane handles 4 channels (head = lane>>3)
__global__ void edge_agg4(const int* __restrict__ src, const int* __restrict__ dst,
                          const float* __restrict__ als, const float* __restrict__ ald,
                          const unsigned* __restrict__ mmax, const float* __restrict__ ssum,
                          const float* __restrict__ H, float* __restrict__ agg,
                          int E, int Et) {
    const long long gid = (long long)blockIdx.x * blockDim.x + threadIdx.x;
    const int e = (int)(gid >> 5);
    const int lane = (int)(gid & 31);
    if (e >= Et) return;
    int s, d; edge_sd(src, dst, e, E, s, d);
    const int hh = lane >> 3;
    const float ev = lrelu(als[4 * s + hh] + ald[4 * d + hh]);
    const float alpha = expf(ev - unford(mmax[4 * d + hh])) / (ssum[4 * d + hh] + 1e-16f);
    const int c0 = lane << 2;
    const float4 hv = *(const float4*)(H + (size_t)s * C128 + c0);
    float* op = agg + (size_t)d * C128 + c0;
    atomicAdd(op + 0, alpha * hv.x);
    atomicAdd(op + 1, alpha * hv.y);
    atomicAdd(op + 2, alpha * hv.z);
    atomicAdd(op + 3, alpha * hv.w);
}

// epilogue: x_next = elu(agg + b)
__global__ void bias_elu128(const float* __restrict__ agg, const float* __restrict__ b,
                            float* __restrict__ out, int N) {
    const long long idx = (long long)blockIdx.x * blockDim.x + threadIdx.x;
    if (idx >= (long long)N * C128) return;
    const float v = agg[idx] + b[(int)(idx & 127)];
    out[idx] = v > 0.f ? v : expm1f(v);
}

// ---------------------------------------------------------------------------
// final layer: 128 -> 6, 1 head
// ---------------------------------------------------------------------------
__global__ void gemm_out6(const float* __restrict__ X, const float* __restrict__ W,
                          float* __restrict__ H, int N) {
    const int idx = blockIdx.x * blockDim.x + threadIdx.x;
    if (idx >= N * 6) return;
    const int n = idx / 6, j = idx - n * 6;
    const float* xr = X + (size_t)n * C128;
    float acc = 0.f;
#pragma unroll 8
    for (int k = 0; k < C128; ++k) acc += xr[k] * W[k * 6 + j];
    H[idx] = acc;
}

__global__ void attn_logits1(const float* __restrict__ H, const float* __restrict__ a_src,
                             const float* __restrict__ a_dst,
                             float* __restrict__ als, float* __restrict__ ald, int N) {
    const int n = blockIdx.x * blockDim.x + threadIdx.x;
    if (n >= N) return;
    float s0 = 0.f, s1 = 0.f;
#pragma unroll
    for (int j = 0; j < 6; ++j) { const float v = H[n * 6 + j]; s0 += v * a_src[j]; s1 += v * a_dst[j]; }
    als[n] = s0; ald[n] = s1;
}

__global__ void edge_max1(const int* __restrict__ src, const int* __restrict__ dst,
                          const float* __restrict__ als, const float* __restrict__ ald,
                          unsigned* __restrict__ mmax, int E, int Et) {
    const int e = blockIdx.x * blockDim.x + threadIdx.x;
    if (e >= Et) return;
    int s, d; edge_sd(src, dst, e, E, s, d);
    atomicMax(&mmax[d], ford(lrelu(als[s] + ald[d])));
}

__global__ void edge_sum1(const int* __restrict__ src, const int* __restrict__ dst,
                          const float* __restrict__ als, const float* __restrict__ ald,
                          const unsigned* __restrict__ mmax, float* __restrict__ ssum,
                          int E, int Et) {
    const int e = blockIdx.x * blockDim.x + threadIdx.x;
    if (e >= Et) return;
    int s, d; edge_sd(src, dst, e, E, s, d);
    atomicAdd(&ssum[d], expf(lrelu(als[s] + ald[d]) - unford(mmax[d])));
}

__global__ void edge_agg1(const int* __restrict__ src, const int* __restrict__ dst,
                          const float* __restrict__ als, const float* __restrict__ ald,
                          const unsigned* __restrict__ mmax, const float* __restrict__ ssum,
                          const float* __restrict__ H, float* __restrict__ agg,
                          int E, int Et) {
    const int e = blockIdx.x * blockDim.x + threadIdx.x;
    if (e >= Et) return;
    int s, d; edge_sd(src, dst, e, E, s, d);
    const float alpha = expf(lrelu(als[s] + ald[d]) - unford(mmax[d])) / (ssum[d] + 1e-16f);
    const float* hp = H + (size_t)s * 6;
    float* op = agg + (size_t)d * 6;
#pragma unroll
    for (int j = 0; j < 6; ++j) atomicAdd(op + j, alpha * hp[j]);
}

__global__ void bias_out6(const float* __restrict__ agg, const float* __restrict__ b,
                          float* __restrict__ out, int N) {
    const int idx = blockIdx.x * blockDim.x + threadIdx.x;
    if (idx >= N * 6) return;
    const int j = idx % 6;
    out[idx] = agg[idx] + b[j];
}

// ---------------------------------------------------------------------------
static inline int cdiv_i(long long a, int b) { return (int)((a + b - 1) / b); }

extern "C" void kernel_launch(void* const* d_in, const int* in_sizes, int n_in,
                              void* d_out, int out_size, void* d_ws, size_t ws_size,
                              hipStream_t stream) {
    const float* x   = (const float*)d_in[0];
    const int*   ei  = (const int*)d_in[1];
    const int N  = in_sizes[2];          // batch[] has N elements
    const int E  = in_sizes[1] / 2;
    const int Et = E + N;                // + self-loops
    const int* srcI = ei;
    const int* dstI = ei + E;

    const float* W[4]  = {(const float*)d_in[3],  (const float*)d_in[7],
                          (const float*)d_in[11], (const float*)d_in[15]};
    const float* AS[4] = {(const float*)d_in[4],  (const float*)d_in[8],
                          (const float*)d_in[12], (const float*)d_in[16]};
    const float* AD[4] = {(const float*)d_in[5],  (const float*)d_in[9],
                          (const float*)d_in[13], (const float*)d_in[17]};
    const float* B[4]  = {(const float*)d_in[6],  (const float*)d_in[10],
                          (const float*)d_in[14], (const float*)d_in[18]};

    // workspace layout (floats): xa[N*128] | h[N*128] | agg[N*128] | als[N*4] | ald[N*4] | ssum[N*4] | mmax[N*4]
    float* ws   = (float*)d_ws;
    float* xa   = ws;
    float* hbuf = xa   + (size_t)N * C128;
    float* agg  = hbuf + (size_t)N * C128;
    float* als  = agg  + (size_t)N * C128;
    float* ald  = als  + (size_t)N * 4;
    float* ssum = ald  + (size_t)N * 4;
    unsigned* mmax = (unsigned*)(ssum + (size_t)N * 4);

    const dim3 blk(256);
    const int gE   = cdiv_i(Et, 256);
    const int gE32 = cdiv_i((long long)Et * 32, 256);

    // ---- layers 0..2 : K -> 128, 4 heads, concat, ELU -------------------
    for (int L = 0; L < 3; ++L) {
        const float* xin = (L == 0) ? x : xa;
        const int K = (L == 0) ? 64 : 128;

        gat_gemm128_wmma<<<cdiv_i(N, 16), blk, 0, stream>>>(xin, W[L], hbuf, N, K);
        attn_logits4<<<cdiv_i((long long)N * 4, 256), blk, 0, stream>>>(hbuf, AS[L], AD[L], als, ald, N);

        hipMemsetAsync(mmax, 0, (size_t)N * 4 * sizeof(unsigned), stream); // ford(-inf)-ish floor
        hipMemsetAsync(ssum, 0, (size_t)N * 4 * sizeof(float), stream);
        hipMemsetAsync(agg,  0, (size_t)N * C128 * sizeof(float), stream);

        edge_max4<<<gE,   blk, 0, stream>>>(srcI, dstI, als, ald, mmax, E, Et);
        edge_sum4<<<gE,   blk, 0, stream>>>(srcI, dstI, als, ald, mmax, ssum, E, Et);
        edge_agg4<<<gE32, blk, 0, stream>>>(srcI, dstI, als, ald, mmax, ssum, hbuf, agg, E, Et);

        bias_elu128<<<cdiv_i((long long)N * C128, 256), blk, 0, stream>>>(agg, B[L], xa, N);
    }

    // ---- final layer : 128 -> 6, 1 head, no activation ------------------
    gemm_out6<<<cdiv_i((long long)N * 6, 256), blk, 0, stream>>>(xa, W[3], hbuf, N);
    attn_logits1<<<cdiv_i(N, 256), blk, 0, stream>>>(hbuf, AS[3], AD[3], als, ald, N);

    hipMemsetAsync(mmax, 0, (size_t)N * sizeof(unsigned), stream);
    hipMemsetAsync(ssum, 0, (size_t)N * sizeof(float), stream);
    hipMemsetAsync(agg,  0, (size_t)N * 6 * sizeof(float), stream);

    edge_max1<<<gE, blk, 0, stream>>>(srcI, dstI, als, ald, mmax, E, Et);
    edge_sum1<<<gE, blk, 0, stream>>>(srcI, dstI, als, ald, mmax, ssum, E, Et);
    edge_agg1<<<gE, blk, 0, stream>>>(srcI, dstI, als, ald, mmax, ssum, hbuf, agg, E, Et);

    bias_out6<<<cdiv_i((long long)N * 6, 256), blk, 0, stream>>>(agg, B[3], (float*)d_out, N);
}